// PatchedInternLMAttention_15307263442949
// MI455X (gfx1250) — compile-verified
//
#include <hip/hip_runtime.h>
#include <hip/hip_bf16.h>
#include <math.h>

// Problem constants (match reference)
constexpr int kB  = 8;     // batch
constexpr int kH  = 16;    // heads
constexpr int kHD = 128;   // head dim
constexpr int kD  = 2048;  // hidden
constexpr int kBS = 64;    // kv block size
constexpr int kMB = 64;    // max blocks per seq

typedef __attribute__((ext_vector_type(2))) float v2f;
typedef __attribute__((ext_vector_type(4))) float v4f;
typedef __attribute__((ext_vector_type(8))) float v8f;

// GEMM tiling
constexpr int TN  = 128;       // N columns per workgroup (8 wmma tiles, 1/wave)
constexpr int KC  = 64;        // K chunk staged in LDS
constexpr int KCP = KC + 4;    // padded LDS row stride (floats): 272B, 16B-aligned,
                               // 4-bank shift per row -> conflict-free b64 frags

// ---------------------------------------------------------------------------
// GEMM: out[M,N] = X[M,K] @ W[N,K]^T + bias[N], M <= 16 (padded to WMMA tile).
// 256 threads = 8 wave32s; each wave owns one 16x16 output tile.
// Per K-chunk: cooperative coalesced float4 global->LDS staging of X and W,
// then V_WMMA_F32_16X16X4_F32 with both fragments from LDS (ds_load_b64).
//
// f32 WMMA VGPR layouts (ISA 7.12.2):
//   A 16x4 : lanes 0-15 -> M=lane, v0=K+0, v1=K+1 ; lanes 16-31 -> K+2/K+3
//   B 4x16 : lanes 0-15 -> N=lane, v0=K+0, v1=K+1 ; lanes 16-31 -> K+2/K+3
//   C/D    : VGPR r, lanes 0-15 -> (M=r, N=lane); lanes 16-31 -> (M=r+8, N=lane-16)
// ---------------------------------------------------------------------------
__global__ __launch_bounds__(256) void gemm_xwt_wmma(
    const float* __restrict__ X, int Mreal, int K,
    const float* __restrict__ W, const float* __restrict__ bias,
    float* __restrict__ out, int ldout)
{
  __shared__ float Xs[16 * KCP];   // 16 x KC tile of X (rows >= Mreal zeroed)
  __shared__ float Ws[TN * KCP];   // TN x KC tile of W

  const int tid  = threadIdx.x;
  const int wave = tid >> 5;       // 0..7
  const int lane = tid & 31;
  const int hi   = lane >> 4;      // 0: K+0/K+1 half, 1: K+2/K+3 half
  const int l    = lane & 15;      // row (A) / col (B) within tile
  const int n0   = blockIdx.x * TN;
  const int nw   = n0 + wave * 16; // this wave's N base

  v8f acc = {};
  for (int k0 = 0; k0 < K; k0 += KC) {
    // ---- stage X tile (16 x KC): 256 v4f, 1 per thread; zero pad rows
    {
      const int row = tid / (KC / 4);          // 0..15
      const int c4  = tid % (KC / 4);
      v4f val = {0.f, 0.f, 0.f, 0.f};
      if (row < Mreal)
        val = *(const v4f*)(X + (size_t)row * K + k0 + c4 * 4);
      *(v4f*)(Xs + row * KCP + c4 * 4) = val;
    }
    // ---- stage W tile (TN x KC): 2048 v4f, 8 per thread, fully coalesced
#pragma unroll
    for (int i = 0; i < TN * (KC / 4) / 256; ++i) {
      const int idx = tid + i * 256;
      const int row = idx / (KC / 4);
      const int c4  = idx % (KC / 4);
      *(v4f*)(Ws + row * KCP + c4 * 4) =
          *(const v4f*)(W + (size_t)(n0 + row) * K + k0 + c4 * 4);
    }
    __syncthreads();

    // prefetch next W chunk while we compute on this one
    if (k0 + KC < K)
      __builtin_prefetch(W + (size_t)(n0 + (tid >> 1)) * K + k0 + KC +
                             (tid & 1) * 32, 0, 0);

    // ---- 16 wmma steps over the chunk, all operands from LDS
    const float* xr = Xs + l * KCP;
    const float* wr = Ws + (wave * 16 + l) * KCP;
#pragma unroll
    for (int k = 0; k < KC; k += 4) {
      const int ka = k + 2 * hi;
      const v2f a = *(const v2f*)(xr + ka);
      const v2f b = *(const v2f*)(wr + ka);
      acc = __builtin_amdgcn_wmma_f32_16x16x4_f32(
          /*neg_a=*/false, a, /*neg_b=*/false, b,
          /*c_mod=*/(short)0, acc, /*reuse_a=*/false, /*reuse_b=*/false);
    }
    __syncthreads();
  }

  const float bv = bias[nw + l];
#pragma unroll
  for (int r = 0; r < 8; ++r) {
    const int mm = r + 8 * hi;
    if (mm < Mreal) out[(size_t)mm * ldout + nw + l] = acc[r] + bv;
  }
}

// ---------------------------------------------------------------------------
// RoPE in-place on q and k: one block per (b,h), 128 threads (one per d).
// ---------------------------------------------------------------------------
__global__ __launch_bounds__(kHD) void rope_kernel(
    float* __restrict__ q, float* __restrict__ k,
    const int* __restrict__ hist)
{
  const int bh = blockIdx.x;
  const int b  = bh / kH;
  const int d  = threadIdx.x;          // 0..127
  const int i  = d & 63;               // frequency index
  const float pos = (float)hist[b];
  const float inv = __powf(10000.0f, -(float)(2 * i) / (float)kHD);
  const float f   = pos * inv;
  const float c = __cosf(f), s = __sinf(f);

  float* qp = q + (size_t)bh * kHD;
  float* kp = k + (size_t)bh * kHD;
  const int other = (d < 64) ? d + 64 : d - 64;
  const float sgn = (d < 64) ? -1.0f : 1.0f;

  const float qd = qp[d], qo = qp[other];
  const float kd = kp[d], ko = kp[other];
  __syncthreads();                      // everyone read before anyone writes
  qp[d] = qd * c + sgn * qo * s;
  kp[d] = kd * c + sgn * ko * s;
}

// ---------------------------------------------------------------------------
// Flash-decode attention: one workgroup per (b,h), 8 wave32s.
// Each wave owns positions s = wave, wave+8, ... <= pos (s==pos -> new token).
// Lane holds q as a v4f (128-bit coalesced KV loads, non-temporal since the
// KV stream is single-pass), dot via shfl_xor reduction, online softmax in
// registers, cross-wave merge in LDS.
// ---------------------------------------------------------------------------
__global__ __launch_bounds__(256) void attn_decode(
    const float* __restrict__ q,      // [B,H,HD] (roped)
    const float* __restrict__ knew,   // [B,H,HD] (roped)
    const float* __restrict__ vnew,   // [B,H,HD]
    const float* __restrict__ k_cache,// [NB,BS,H,HD]
    const float* __restrict__ v_cache,
    const int*   __restrict__ hist,   // [B]
    const int*   __restrict__ blkoff, // [B,MB]
    float* __restrict__ out)          // [B,H,HD]
{
  const int bh   = blockIdx.x;
  const int b    = bh / kH;
  const int h    = bh % kH;
  const int pos  = hist[b];
  const int wave = threadIdx.x >> 5;   // 0..7
  const int lane = threadIdx.x & 31;   // 0..31

  const v4f qv = ((const v4f*)(q + (size_t)bh * kHD))[lane];
  const float scale = 0.08838834764831845f;  // 1/sqrt(128)

  float m = -INFINITY, l = 0.0f;
  v4f acc = {0.f, 0.f, 0.f, 0.f};

  for (int s = wave; s <= pos; s += 8) {
    const float* kp;
    const float* vp;
    if (s == pos) {
      kp = knew + (size_t)bh * kHD;
      vp = vnew + (size_t)bh * kHD;
    } else {
      const int blk = blkoff[b * kMB + (s >> 6)];
      const size_t row = ((size_t)blk * kBS + (s & 63)) * kH + h;
      kp = k_cache + row * kHD;
      vp = v_cache + row * kHD;
    }
    const v4f kv = __builtin_nontemporal_load((const v4f*)kp + lane);
    float d = qv.x * kv.x + qv.y * kv.y + qv.z * kv.z + qv.w * kv.w;
#pragma unroll
    for (int off = 16; off > 0; off >>= 1) d += __shfl_xor(d, off, 32);
    d *= scale;

    const float mnew = fmaxf(m, d);
    const float corr = __expf(m - mnew);   // m==-inf, mnew finite -> 0
    const float p    = __expf(d - mnew);
    const v4f vv     = __builtin_nontemporal_load((const v4f*)vp + lane);
    acc.x = acc.x * corr + p * vv.x;
    acc.y = acc.y * corr + p * vv.y;
    acc.z = acc.z * corr + p * vv.z;
    acc.w = acc.w * corr + p * vv.w;
    l = l * corr + p;
    m = mnew;
  }

  __shared__ float sm[8];
  __shared__ float sl[8];
  __shared__ float sacc[8][kHD];
  if (lane == 0) { sm[wave] = m; sl[wave] = l; }
  ((v4f*)sacc[wave])[lane] = acc;
  __syncthreads();

  if (wave == 0) {
    float M = -INFINITY;
#pragma unroll
    for (int w = 0; w < 8; ++w) M = fmaxf(M, sm[w]);
    float L = 0.0f;
    v4f o = {0.f, 0.f, 0.f, 0.f};
#pragma unroll
    for (int w = 0; w < 8; ++w) {
      const float c = __expf(sm[w] - M);   // waves with no work: exp(-inf)=0
      L += sl[w] * c;
      const v4f a = ((const v4f*)sacc[w])[lane];
      o.x += c * a.x; o.y += c * a.y; o.z += c * a.z; o.w += c * a.w;
    }
    const float invL = 1.0f / L;
    v4f r = {o.x * invL, o.y * invL, o.z * invL, o.w * invL};
    ((v4f*)(out + (size_t)bh * kHD))[lane] = r;
  }
}

// ---------------------------------------------------------------------------
// Host launcher.  Input order: hidden_states, k_cache, v_cache, Wq, bq, Wk,
// bk, Wv, bv, Wo, bo, history_lengths, block_offsets.
// ---------------------------------------------------------------------------
extern "C" void kernel_launch(void* const* d_in, const int* in_sizes, int n_in,
                              void* d_out, int out_size, void* d_ws, size_t ws_size,
                              hipStream_t stream) {
  (void)in_sizes; (void)n_in; (void)out_size; (void)ws_size;
  const float* hs      = (const float*)d_in[0];
  const float* k_cache = (const float*)d_in[1];
  const float* v_cache = (const float*)d_in[2];
  const float* Wq      = (const float*)d_in[3];
  const float* bq      = (const float*)d_in[4];
  const float* Wk      = (const float*)d_in[5];
  const float* bk      = (const float*)d_in[6];
  const float* Wv      = (const float*)d_in[7];
  const float* bv      = (const float*)d_in[8];
  const float* Wo      = (const float*)d_in[9];
  const float* bo      = (const float*)d_in[10];
  const int*   hist    = (const int*)d_in[11];
  const int*   blkoff  = (const int*)d_in[12];

  float* ws = (float*)d_ws;
  float* qw = ws;                    // [B,D]
  float* kw = ws + (size_t)kB * kD;  // [B,D]
  float* vw = ws + (size_t)2 * kB * kD;
  float* aw = ws + (size_t)3 * kB * kD;

  const dim3 gtiles(kD / TN);        // 16 workgroups per GEMM
  gemm_xwt_wmma<<<gtiles, 256, 0, stream>>>(hs, kB, kD, Wq, bq, qw, kD);
  gemm_xwt_wmma<<<gtiles, 256, 0, stream>>>(hs, kB, kD, Wk, bk, kw, kD);
  gemm_xwt_wmma<<<gtiles, 256, 0, stream>>>(hs, kB, kD, Wv, bv, vw, kD);

  rope_kernel<<<kB * kH, kHD, 0, stream>>>(qw, kw, hist);

  attn_decode<<<kB * kH, 256, 0, stream>>>(qw, kw, vw, k_cache, v_cache,
                                           hist, blkoff, aw);

  gemm_xwt_wmma<<<gtiles, 256, 0, stream>>>(aw, kB, kD, Wo, bo,
                                           (float*)d_out, kD);
}